// TransformerEdgePolicy_10445360464470
// MI455X (gfx1250) — compile-verified
//
#include <hip/hip_runtime.h>
#include <hip/hip_bf16.h>

// S=64 N=32 E=256 H=8 D=32 DFF=1024 B=N*H=256, DEG=8 top-k, fp32 throughout.
#define S_DIM   64
#define N_DIM   32
#define E_DIM   256
#define H_DIM   8
#define D_DIM   32
#define B_DIM   256
#define DFF_DIM 1024
#define DEG     8
#define LN_EPS  1e-5f
#define SCALING 0.17677669529663687f   // D^-0.5 = 1/sqrt(32)

typedef float v2f __attribute__((ext_vector_type(2)));
typedef float v8f __attribute__((ext_vector_type(8)));

// ---------------------------------------------------------------------------
// WMMA f32 GEMM:  C[M,Ncol] = act(A[M,K] @ W[Ncol,K]^T + bias[Ncol])
// Register-blocked 32x32 tile per wave: 2 A-frags x 2 B-frags -> 4 WMMAs per
// k-step of V_WMMA_F32_16X16X4_F32 (1 fragment load per WMMA instead of 2).
// Fragment layout (ISA 7.12.2), lane=(half,r):
//   A 16x4 frag:  a[j] = A[rowBase + r, k0 + 2*half + j]
//   B 4x16 frag:  b[j] = W[colBase + r, k0 + 2*half + j]
//   C 16x16 frag: VGPR i -> row rowBase + i + 8*half, col colBase + r
// ---------------------------------------------------------------------------
__global__ void gemm_wmma_f32(const float* __restrict__ A,
                              const float* __restrict__ W,
                              const float* __restrict__ bias,
                              float* __restrict__ C,
                              int M, int Ncol, int K, int relu)
{
    const int wavesPerBlock = blockDim.x >> 5;
    const int wave = blockIdx.x * wavesPerBlock + ((int)threadIdx.x >> 5);
    const int lane = threadIdx.x & 31;
    const int tilesN = Ncol >> 5;                 // 32-wide tiles
    const int totalTiles = (M >> 5) * tilesN;
    if (wave >= totalTiles) return;               // wave-uniform (EXEC all-1s)

    const int tileM = (wave / tilesN) << 5;
    const int tileN = (wave % tilesN) << 5;
    const int r    = lane & 15;
    const int half = lane >> 4;

    const float* arow0 = A + (size_t)(tileM + r)      * K + 2 * half;
    const float* arow1 = A + (size_t)(tileM + 16 + r) * K + 2 * half;
    const float* brow0 = W + (size_t)(tileN + r)      * K + 2 * half;
    const float* brow1 = W + (size_t)(tileN + 16 + r) * K + 2 * half;

    v8f acc00 = {}, acc01 = {}, acc10 = {}, acc11 = {};
    #pragma unroll 4
    for (int k0 = 0; k0 < K; k0 += 4) {
        v2f a0 = *(const v2f*)(arow0 + k0);
        v2f a1 = *(const v2f*)(arow1 + k0);
        v2f b0 = *(const v2f*)(brow0 + k0);
        v2f b1 = *(const v2f*)(brow1 + k0);
        acc00 = __builtin_amdgcn_wmma_f32_16x16x4_f32(false, a0, false, b0, (short)0, acc00, false, false);
        acc01 = __builtin_amdgcn_wmma_f32_16x16x4_f32(false, a0, false, b1, (short)0, acc01, false, false);
        acc10 = __builtin_amdgcn_wmma_f32_16x16x4_f32(false, a1, false, b0, (short)0, acc10, false, false);
        acc11 = __builtin_amdgcn_wmma_f32_16x16x4_f32(false, a1, false, b1, (short)0, acc11, false, false);
    }

    const int col0 = tileN + r;
    const int col1 = tileN + 16 + r;
    const float bb0 = bias ? bias[col0] : 0.f;
    const float bb1 = bias ? bias[col1] : 0.f;
    #pragma unroll
    for (int i = 0; i < 8; ++i) {
        const size_t row0 = (size_t)(tileM + i + 8 * half) * Ncol;
        const size_t row1 = (size_t)(tileM + 16 + i + 8 * half) * Ncol;
        float v00 = acc00[i] + bb0, v01 = acc01[i] + bb1;
        float v10 = acc10[i] + bb0, v11 = acc11[i] + bb1;
        if (relu) {
            v00 = fmaxf(v00, 0.f); v01 = fmaxf(v01, 0.f);
            v10 = fmaxf(v10, 0.f); v11 = fmaxf(v11, 0.f);
        }
        C[row0 + col0] = v00;  C[row0 + col1] = v01;
        C[row1 + col0] = v10;  C[row1 + col1] = v11;
    }
}

// ---------------------------------------------------------------------------
// Attention scores + edge bias + hard top-8 + softmax, emitting a COMPACTED
// (index, weight) list so the value pass only touches 8/64 of edge_v.
// One wave32 per (b,s) query row; each lane owns keys t=lane and t=lane+32.
// ---------------------------------------------------------------------------
__global__ void attn_topk_softmax(const float* __restrict__ QK,      // (S*N, 2E)
                                  const float* __restrict__ edge_k,  // (S*S, N, E)
                                  int*   __restrict__ topk_idx,      // (B*S, DEG)
                                  float* __restrict__ topk_w)        // (B*S, DEG)
{
    const int wave = blockIdx.x * (blockDim.x >> 5) + ((int)threadIdx.x >> 5);
    const int lane = threadIdx.x & 31;
    if (wave >= B_DIM * S_DIM) return;
    const int b = wave >> 6;          // / S
    const int s = wave & 63;
    const int n = b >> 3;             // / H
    const int h = b & 7;

    // Prefetch this wave's second-half edge_k row (streams the 134MB tensor).
    __builtin_prefetch(edge_k + ((size_t)((s * S_DIM + lane + 32) * N_DIM + n)) * E_DIM
                              + h * D_DIM, 0, 1);

    // q row is wave-uniform: QK[(s*N+n)*2E + h*D + 0..31]
    const float* qptr = QK + (size_t)(s * N_DIM + n) * (2 * E_DIM) + h * D_DIM;
    float4 q4[8];
    #pragma unroll
    for (int j = 0; j < 8; ++j) q4[j] = *(const float4*)(qptr + 4 * j);

    float att[2];
    #pragma unroll
    for (int ti = 0; ti < 2; ++ti) {
        const int t = lane + 32 * ti;
        const float* kptr  = QK + (size_t)(t * N_DIM + n) * (2 * E_DIM) + E_DIM + h * D_DIM;
        const float* ekptr = edge_k + ((size_t)((s * S_DIM + t) * N_DIM + n)) * E_DIM + h * D_DIM;
        float acc = 0.f;
        #pragma unroll
        for (int j = 0; j < 8; ++j) {
            float4 kv = *(const float4*)(kptr + 4 * j);
            float4 ev = *(const float4*)(ekptr + 4 * j);
            acc += q4[j].x * (kv.x + ev.x) + q4[j].y * (kv.y + ev.y)
                 + q4[j].z * (kv.z + ev.z) + q4[j].w * (kv.w + ev.w);
        }
        att[ti] = acc * SCALING;      // scaling folded out of q
    }

    // Iteratively extract 8 maxima; kth = 8th largest, maxv = largest.
    float w0 = att[0], w1 = att[1];
    float kth = 0.f, maxv = 0.f;
    #pragma unroll
    for (int i = 0; i < DEG; ++i) {
        float m = fmaxf(w0, w1);
        for (int off = 16; off > 0; off >>= 1)
            m = fmaxf(m, __shfl_xor(m, off, 32));
        if (i == 0) maxv = m;
        kth = m;
        if (w0 == m)      w0 = -__builtin_inff();
        else if (w1 == m) w1 = -__builtin_inff();
    }

    const bool keep0 = att[0] >= kth;
    const bool keep1 = att[1] >= kth;
    float e0 = keep0 ? __expf(att[0] - maxv) : 0.f;
    float e1 = keep1 ? __expf(att[1] - maxv) : 0.f;
    float ssum = e0 + e1;
    for (int off = 16; off > 0; off >>= 1) ssum += __shfl_xor(ssum, off, 32);
    const float inv = 1.f / ssum;

    // Compact kept entries with wave32 ballot + prefix popcount.
    const unsigned m0 = __builtin_amdgcn_ballot_w32(keep0);
    const unsigned m1 = __builtin_amdgcn_ballot_w32(keep1);
    const unsigned below = (lane == 0) ? 0u : (0xFFFFFFFFu >> (32 - lane));
    const int base = wave * DEG;
    if (keep0) {
        int pos = __popc(m0 & below);
        if (pos < DEG) { topk_idx[base + pos] = lane;      topk_w[base + pos] = e0 * inv; }
    }
    if (keep1) {
        int pos = __popc(m0) + __popc(m1 & below);
        if (pos < DEG) { topk_idx[base + pos] = lane + 32; topk_w[base + pos] = e1 * inv; }
    }
}

// ---------------------------------------------------------------------------
// Sparse value gather: out[b,s,d] = sum_j w_j * (v[b,t_j,d] + ev[a,t_j,d]).
// Reads only DEG=8 of 64 edge_v rows per query (134MB -> 16.8MB traffic).
// One wave per (b,s); lane = d. Output written directly in (S,N,E) layout.
// ---------------------------------------------------------------------------
__global__ void gather_value(const float* __restrict__ Vbuf,    // (S*N, E)
                             const float* __restrict__ edge_v,  // flat, (B*S, S, D) view
                             const int*   __restrict__ topk_idx,
                             const float* __restrict__ topk_w,
                             float* __restrict__ Out)            // (S*N, E)
{
    const int wave = blockIdx.x * (blockDim.x >> 5) + ((int)threadIdx.x >> 5);
    const int lane = threadIdx.x & 31;
    if (wave >= B_DIM * S_DIM) return;
    const int b = wave >> 6, s = wave & 63;
    const int n = b >> 3,    h = b & 7;

    float acc = 0.f;
    #pragma unroll
    for (int j = 0; j < DEG; ++j) {
        const int   t  = topk_idx[wave * DEG + j];
        const float wv = topk_w[wave * DEG + j];
        const float vv = Vbuf[(size_t)(t * N_DIM + n) * E_DIM + h * D_DIM + lane];
        const float ev = edge_v[((size_t)wave * S_DIM + t) * D_DIM + lane];
        acc += wv * (vv + ev);
    }
    Out[(size_t)(s * N_DIM + n) * E_DIM + h * D_DIM + lane] = acc;
}

// ---------------------------------------------------------------------------
// Out = LayerNorm(X + Y) * gamma + beta, one wave32 per row of E=256.
// ---------------------------------------------------------------------------
__global__ void add_layernorm(const float* __restrict__ X,
                              const float* __restrict__ Y,
                              const float* __restrict__ gamma,
                              const float* __restrict__ beta,
                              float* __restrict__ Out, int rows)
{
    const int wave = blockIdx.x * (blockDim.x >> 5) + ((int)threadIdx.x >> 5);
    const int lane = threadIdx.x & 31;
    if (wave >= rows) return;
    const float* xr = X + (size_t)wave * E_DIM;
    const float* yr = Y + (size_t)wave * E_DIM;

    float v[8];
    float sum = 0.f;
    #pragma unroll
    for (int j = 0; j < 8; ++j) { v[j] = xr[j * 32 + lane] + yr[j * 32 + lane]; sum += v[j]; }
    for (int off = 16; off > 0; off >>= 1) sum += __shfl_xor(sum, off, 32);
    const float mean = sum * (1.f / E_DIM);

    float var = 0.f;
    #pragma unroll
    for (int j = 0; j < 8; ++j) { float d = v[j] - mean; var += d * d; }
    for (int off = 16; off > 0; off >>= 1) var += __shfl_xor(var, off, 32);
    const float rstd = rsqrtf(var * (1.f / E_DIM) + LN_EPS);

    float* outr = Out + (size_t)wave * E_DIM;
    #pragma unroll
    for (int j = 0; j < 8; ++j) {
        const int c = j * 32 + lane;
        outr[c] = (v[j] - mean) * rstd * gamma[c] + beta[c];
    }
}

// ---------------------------------------------------------------------------
extern "C" void kernel_launch(void* const* d_in, const int* in_sizes, int n_in,
                              void* d_out, int out_size, void* d_ws, size_t ws_size,
                              hipStream_t stream)
{
    const float* src       = (const float*)d_in[0];
    const float* edge_k    = (const float*)d_in[1];
    const float* edge_v    = (const float*)d_in[2];
    const float* in_proj_w = (const float*)d_in[3];
    const float* in_proj_b = (const float*)d_in[4];
    const float* v_w       = (const float*)d_in[5];
    const float* v_b       = (const float*)d_in[6];
    const float* out_w     = (const float*)d_in[7];
    const float* out_b     = (const float*)d_in[8];
    const float* w1        = (const float*)d_in[9];
    const float* b1        = (const float*)d_in[10];
    const float* w2        = (const float*)d_in[11];
    const float* b2        = (const float*)d_in[12];
    const float* g1        = (const float*)d_in[13];
    const float* be1       = (const float*)d_in[14];
    const float* g2        = (const float*)d_in[15];
    const float* be2       = (const float*)d_in[16];

    const int ROWS = S_DIM * N_DIM;                 // 2048
    float* ws = (float*)d_ws;
    float* QK   = ws;  ws += (size_t)ROWS * 2 * E_DIM;      // 1,048,576 f
    float* Vb   = ws;  ws += (size_t)ROWS * E_DIM;          //   524,288 f
    float* AOut = ws;  ws += (size_t)ROWS * E_DIM;
    float* A2   = ws;  ws += (size_t)ROWS * E_DIM;
    float* Xb   = ws;  ws += (size_t)ROWS * E_DIM;
    float* FF1  = ws;  ws += (size_t)ROWS * DFF_DIM;        // 2,097,152 f
    float* FF2  = ws;  ws += (size_t)ROWS * E_DIM;
    float* TW   = ws;  ws += (size_t)B_DIM * S_DIM * DEG;   //   131,072 f
    int*   TI   = (int*)ws;                                 //   131,072 i
    // total ~= 24.1 MB of workspace

    const dim3 blk(128);                                     // 4 waves / block
    // 32x32 tiles, one wave each, 4 waves per block. All dims are x32 multiples.
    auto gemmBlocks = [](int M, int Nc) { return dim3(((M >> 5) * (Nc >> 5)) / 4); };

    // 1) fused q/k projection
    gemm_wmma_f32<<<gemmBlocks(ROWS, 2 * E_DIM), blk, 0, stream>>>(
        src, in_proj_w, in_proj_b, QK, ROWS, 2 * E_DIM, E_DIM, 0);
    // 2) value projection
    gemm_wmma_f32<<<gemmBlocks(ROWS, E_DIM), blk, 0, stream>>>(
        src, v_w, v_b, Vb, ROWS, E_DIM, E_DIM, 0);
    // 3) scores + edge bias + top-8 + softmax (compacted)
    attn_topk_softmax<<<dim3((B_DIM * S_DIM) / 4), blk, 0, stream>>>(QK, edge_k, TI, TW);
    // 4) sparse value gather (+edge_v)
    gather_value<<<dim3((B_DIM * S_DIM) / 4), blk, 0, stream>>>(Vb, edge_v, TI, TW, AOut);
    // 5) output projection
    gemm_wmma_f32<<<gemmBlocks(ROWS, E_DIM), blk, 0, stream>>>(
        AOut, out_w, out_b, A2, ROWS, E_DIM, E_DIM, 0);
    // 6) x = LN(src + attn)
    add_layernorm<<<dim3(ROWS / 4), blk, 0, stream>>>(src, A2, g1, be1, Xb, ROWS);
    // 7) FFN up + ReLU
    gemm_wmma_f32<<<gemmBlocks(ROWS, DFF_DIM), blk, 0, stream>>>(
        Xb, w1, b1, FF1, ROWS, DFF_DIM, E_DIM, 1);
    // 8) FFN down
    gemm_wmma_f32<<<gemmBlocks(ROWS, E_DIM), blk, 0, stream>>>(
        FF1, w2, b2, FF2, ROWS, E_DIM, DFF_DIM, 0);
    // 9) out = LN(x + ff)
    add_layernorm<<<dim3(ROWS / 4), blk, 0, stream>>>(Xb, FF2, g2, be2, (float*)d_out, ROWS);
}